// BoundaryLoss_4063039062327
// MI455X (gfx1250) — compile-verified
//
#include <hip/hip_runtime.h>

// Reference dims (setup_inputs): predictions [32,3,768,768] f32, labels [32,768,768] int
#define DIM_B 32
#define DIM_C 3
#define DIM_H 768
#define DIM_W 768
#define PLANE (DIM_H * DIM_W)   // 589824
#define NROWS (DIM_B * DIM_H)   // 24576
#define NBLK  4096
#define TPB   192               // 6 wave32 waves; 192 threads * 4 px = 768 = W

typedef __attribute__((ext_vector_type(2))) float v2f;
typedef __attribute__((ext_vector_type(8))) float v8f;

// Full-precision 32-lane sum. One V_WMMA_F32_16X16X4_F32 does the first
// butterfly step on the matrix unit: A(16x4)=ones, per-lane value v in B's
// first VGPR slot, 0 in the second => D[m][n] = v_n + v_{n+16} for every m,
// independent of the exact K-slot mapping. D's VGPR0 hands lane L the value
// v_{L%16} + v_{L%16+16}; four xor-shuffles finish the tree.
__device__ __forceinline__ float wave_sum32_wmma(float v) {
  v2f a; a.x = 1.0f; a.y = 1.0f;
  v2f b; b.x = v;    b.y = 0.0f;
  v8f c = {};
  c = __builtin_amdgcn_wmma_f32_16x16x4_f32(false, a, false, b, (short)0, c,
                                            false, false);
  float s = c[0];
  s += __shfl_xor(s, 8, 32);
  s += __shfl_xor(s, 4, 32);
  s += __shfl_xor(s, 2, 32);
  s += __shfl_xor(s, 1, 32);
  return s;  // full 32-lane sum in every lane
}

__device__ __forceinline__ float ce3(float a, float b, float c, int lab) {
  float mx  = fmaxf(a, fmaxf(b, c));
  float lse = __logf(__expf(a - mx) + __expf(b - mx) + __expf(c - mx)) + mx;
  float sel = (lab == 0) ? a : ((lab == 1) ? b : c);
  return lse - sel;  // -log_softmax[lab]
}

__global__ __launch_bounds__(TPB) void boundary_ce_partials(
    const float* __restrict__ pred, const int* __restrict__ labels,
    float2* __restrict__ partials) {
  const int t  = threadIdx.x;
  const int x0 = t << 2;  // 4 consecutive pixels per thread
  float num = 0.0f, den = 0.0f;

  for (int row = blockIdx.x; row < NROWS; row += gridDim.x) {
    const int bi = row / DIM_H;
    const int y  = row - bi * DIM_H;
    // np.gradient stencil with edge one-sided differences
    const int yU = (y == 0) ? 0 : y - 1;
    const int yD = (y == DIM_H - 1) ? DIM_H - 1 : y + 1;

    const int* rowC = labels + (size_t)bi * PLANE + (size_t)y  * DIM_W;
    const int* rowU = labels + (size_t)bi * PLANE + (size_t)yU * DIM_W;
    const int* rowD = labels + (size_t)bi * PLANE + (size_t)yD * DIM_W;

    const int4 lc = *(const int4*)(rowC + x0);
    const int4 lu = *(const int4*)(rowU + x0);
    const int4 ld = *(const int4*)(rowD + x0);
    const int labL = (x0 == 0)           ? lc.x : rowC[x0 - 1];
    const int labR = (x0 + 4 >= DIM_W)   ? lc.w : rowC[x0 + 4];

    const float* prow = pred + (size_t)bi * DIM_C * PLANE + (size_t)y * DIM_W + x0;
    const float4 p0 = *(const float4*)(prow);
    const float4 p1 = *(const float4*)(prow + PLANE);
    const float4 p2 = *(const float4*)(prow + 2 * (size_t)PLANE);

    const int   lab[4] = {lc.x, lc.y, lc.z, lc.w};
    const int   xl[4]  = {labL, lc.x, lc.y, lc.z};
    const int   xr[4]  = {lc.y, lc.z, lc.w, labR};
    const int   yu[4]  = {lu.x, lu.y, lu.z, lu.w};
    const int   yd[4]  = {ld.x, ld.y, ld.z, ld.w};
    const float pa[4]  = {p0.x, p0.y, p0.z, p0.w};
    const float pb[4]  = {p1.x, p1.y, p1.z, p1.w};
    const float pc[4]  = {p2.x, p2.y, p2.z, p2.w};

#pragma unroll
    for (int i = 0; i < 4; ++i) {
      // boundary of class-1/class-2 indicator union == "stencil labels differ"
      const float m  = ((yu[i] != yd[i]) || (xl[i] != xr[i])) ? 1.0f : 0.0f;
      const float ce = ce3(pa[i], pb[i], pc[i], lab[i]);
      num = fmaf(ce, m, num);
      den += m;
    }
  }

  // deterministic block reduction: WMMA+shuffle per wave, then serial over waves
  num = wave_sum32_wmma(num);
  den = wave_sum32_wmma(den);

  __shared__ float2 wpart[TPB / 32];
  const int wid = t >> 5;
  if ((t & 31) == 0) wpart[wid] = make_float2(num, den);
  __syncthreads();
  if (t == 0) {
    float n = 0.0f, d = 0.0f;
#pragma unroll
    for (int i = 0; i < TPB / 32; ++i) { n += wpart[i].x; d += wpart[i].y; }
    partials[blockIdx.x] = make_float2(n, d);
  }
}

__global__ __launch_bounds__(256) void finalize_kernel(
    const float2* __restrict__ partials, float* __restrict__ out) {
  __shared__ double sn[256];
  __shared__ double sd[256];
  double n = 0.0, d = 0.0;
  for (int i = threadIdx.x; i < NBLK; i += 256) {
    const float2 p = partials[i];
    n += (double)p.x;
    d += (double)p.y;
  }
  sn[threadIdx.x] = n;
  sd[threadIdx.x] = d;
  __syncthreads();
  for (int off = 128; off > 0; off >>= 1) {
    if ((int)threadIdx.x < off) {
      sn[threadIdx.x] += sn[threadIdx.x + off];
      sd[threadIdx.x] += sd[threadIdx.x + off];
    }
    __syncthreads();
  }
  if (threadIdx.x == 0) out[0] = (float)(sn[0] / (sd[0] + 1e-8));
}

extern "C" void kernel_launch(void* const* d_in, const int* in_sizes, int n_in,
                              void* d_out, int out_size, void* d_ws, size_t ws_size,
                              hipStream_t stream) {
  const float* pred   = (const float*)d_in[0];   // [B,C,H,W] f32
  const int*   labels = (const int*)d_in[1];     // [B,H,W] int
  float*       out    = (float*)d_out;           // scalar
  float2*      part   = (float2*)d_ws;           // NBLK float2 = 32 KB scratch

  boundary_ce_partials<<<NBLK, TPB, 0, stream>>>(pred, labels, part);
  finalize_kernel<<<1, 256, 0, stream>>>(part, out);
}